// CBlock3D_68702296867137
// MI455X (gfx1250) — compile-verified
//
#include <hip/hip_runtime.h>
#include <hip/hip_bf16.h>
#include <math.h>

// ---------------------------------------------------------------------------
// Problem constants (from reference): N=4, CIN=16, COUT=32, K=3, D=16, H=64, W=64
// ---------------------------------------------------------------------------
#define CN    4
#define CIN   16
#define COUT  32
#define DD    16
#define HH    64
#define WW    64
#define NPIX  (CN*DD*HH*WW)          // 262144 output pixels
#define NROW  (NPIX/WW)              // 4096 output rows (n,d,h)
#define NOUT  (NPIX*COUT)            // 8388608 output elements
#define WPCNT (27*COUT*CIN)          // 13824 preprocessed weights
#define RS    ((WW + 2) * CIN)       // 1056: xT row stride (w halo: -1..64)

typedef __attribute__((ext_vector_type(2))) float v2f;
typedef __attribute__((ext_vector_type(8))) float v8f;

// ---------------------------------------------------------------------------
// Kernel 1: weight preprocessing (soft-threshold + Hamming window)
// Wp layout: [tap 0..26][cout 0..31][cin 0..15]  (B-panel friendly)
// ---------------------------------------------------------------------------
__global__ __launch_bounds__(512) void prep_weights(const float* __restrict__ w,
                                                    float* __restrict__ Wp) {
    int t = threadIdx.x;                 // 0..511 -> (cout, cin)
    int co = t >> 4;
    int ci = t & 15;
    const float* src = w + (co * CIN + ci) * 27;   // [cout][cin][27 taps]
    float vals[27];
    float mx = -3.4e38f;
    #pragma unroll
    for (int i = 0; i < 27; ++i) { vals[i] = src[i]; mx = fmaxf(mx, vals[i]); }
    float cutoff = mx * 0.5f;
    float win[3];
    #pragma unroll
    for (int i = 0; i < 3; ++i)
        win[i] = 0.54f - 0.46f * cosf(3.14159265358979323846f * (float)i);
    #pragma unroll
    for (int kd = 0; kd < 3; ++kd)
        #pragma unroll
        for (int kh = 0; kh < 3; ++kh)
            #pragma unroll
            for (int kw = 0; kw < 3; ++kw) {
                int tap = (kd * 3 + kh) * 3 + kw;
                float v = vals[tap];
                float sh = copysignf(fmaxf(fabsf(v) - cutoff * 0.01f, 0.0f), v);
                float o  = (v < cutoff) ? sh : v;
                o *= win[kd] * win[kh] * win[kw];
                Wp[(tap * COUT + co) * CIN + ci] = o;
            }
}

// ---------------------------------------------------------------------------
// Kernel 2: zero the per-channel sum / sumsq accumulators (64 floats)
// ---------------------------------------------------------------------------
__global__ void zero_sums(float* __restrict__ sums) {
    sums[threadIdx.x] = 0.0f;
}

// ---------------------------------------------------------------------------
// Kernel 3: NCDHW -> NDHWC transpose of x with zero w-halo
// One block per (n,d,h): row layout = [w=-1 .. w=64] x 16 cin (1056 floats)
// ---------------------------------------------------------------------------
__global__ __launch_bounds__(256) void transpose_x(const float* __restrict__ x,
                                                   float* __restrict__ xT) {
    __shared__ float t[CIN][WW + 1];
    int g = blockIdx.x;                 // 0 .. 4095
    int h = g & 63;
    int d = (g >> 6) & 15;
    int n = g >> 10;
    float* dst = xT + g * RS;
    {
        int c  = threadIdx.x >> 4;            // 0..15
        int wv = (threadIdx.x & 15) << 2;     // 0,4,..,60
        float4 v = *(const float4*)(x + (((n * CIN + c) * DD + d) << 12) + (h << 6) + wv);
        t[c][wv] = v.x; t[c][wv + 1] = v.y; t[c][wv + 2] = v.z; t[c][wv + 3] = v.w;
    }
    // zero w-halo: pixel -1 (offset 0..15) and pixel 64 (offset 1040..1055)
    if (threadIdx.x < 32) {
        int c = threadIdx.x & 15;
        int side = threadIdx.x >> 4;          // 0 -> w=-1, 1 -> w=64
        dst[side * (WW + 1) * CIN + c] = 0.0f;
    }
    __syncthreads();
    {
        int p  = threadIdx.x >> 2;            // pixel 0..63
        int c0 = (threadIdx.x & 3) << 2;      // 0,4,8,12
        float4 v = make_float4(t[c0][p], t[c0 + 1][p], t[c0 + 2][p], t[c0 + 3][p]);
        *(float4*)(dst + ((p + 1) << 4) + c0) = v;   // +1: skip w=-1 halo
    }
}

// ---------------------------------------------------------------------------
// Kernel 4: implicit-GEMM conv via V_WMMA_F32_16X16X4_F32
// One wave owns a full output row: 64 pixels (4 M-tiles) x 32 couts
// (two N halves) -> 8 persistent f32 accumulators. The w-halo makes every
// A-fragment load unconditional; weights are staged to LDS with the gfx1250
// async global->LDS path (ASYNCcnt) while prefetches cover the x rows.
// ---------------------------------------------------------------------------
#define CONV_BLOCKS (NROW / 8)        // 512 blocks x 8 waves = 1 row per wave

__global__ __launch_bounds__(256) void conv_wmma(const float* __restrict__ xT,
                                                 const float* __restrict__ Wp,
                                                 const float* __restrict__ bias,
                                                 float* __restrict__ y,
                                                 float* __restrict__ g_sums) {
    __shared__ float lWp[WPCNT];         // 55,296 B
    __shared__ float lSum[2 * COUT];     // [0..31] sum, [32..63] sumsq

    int tid   = threadIdx.x;
    int lane  = tid & 31;
    int m     = lane & 15;       // M row within tile / N col (cout)
    int hi    = lane >> 4;       // lane half
    int kfrag = hi << 1;         // k-offset within a K=4 fragment (0 or 2)

    int r = blockIdx.x * 8 + (tid >> 5); // output row id 0..4095
    int h = r & 63;
    int d = (r >> 6) & 15;
    int n = r >> 10;

    // Async-stage the preprocessed weights straight into LDS (b128 per lane,
    // tracked by ASYNCcnt) -- no VGPR round trip.
    {
        unsigned lbase = (unsigned)(uintptr_t)
            (__attribute__((address_space(3))) char*)lWp;
        unsigned long long gbase = (unsigned long long)(uintptr_t)Wp;
        for (int j = tid; j < WPCNT / 4; j += 256) {
            unsigned           laddr = lbase + 16u * (unsigned)j;
            unsigned long long gaddr = gbase + 16ull * (unsigned)j;
            asm volatile("global_load_async_to_lds_b128 %0, %1, off"
                         :: "v"(laddr), "v"(gaddr) : "memory");
        }
    }

    // Prefetch the nine x-rows this wave will consume (global_prefetch_b8),
    // overlapping HBM latency with the async weight staging above.
    for (int kd = 0; kd < 3; ++kd) {
        int dz = d + kd - 1;
        if (dz < 0 || dz >= DD) continue;
        for (int kh = 0; kh < 3; ++kh) {
            int hz = h + kh - 1;
            if (hz < 0 || hz >= HH) continue;
            const float* row = xT + ((n * DD + dz) * HH + hz) * RS;
            __builtin_prefetch(row + (lane << 5), 0, 3);   // 32 lanes x 128B
        }
    }

    if (tid < 2 * COUT) lSum[tid] = 0.0f;
    asm volatile("s_wait_asynccnt 0" ::: "memory");
    __syncthreads();

    v8f acc[2][4];
    #pragma unroll
    for (int u = 0; u < 2; ++u)
        #pragma unroll
        for (int t = 0; t < 4; ++t) acc[u][t] = (v8f){};

    for (int kd = 0; kd < 3; ++kd) {
        int dz = d + kd - 1;
        if (dz < 0 || dz >= DD) continue;
        for (int kh = 0; kh < 3; ++kh) {
            int hz = h + kh - 1;
            if (hz < 0 || hz >= HH) continue;
            const float* row = xT + ((n * DD + dz) * HH + hz) * RS;
            int tapBase = (kd * 3 + kh) * 3;
            #pragma unroll
            for (int kw = 0; kw < 3; ++kw) {
                // (m + kw - 1) + 1 halo pixels -> (m + kw) << 4 ; fold kfrag
                const float* pa  = row + ((m + kw) << 4) + kfrag;
                const float* lB0 = &lWp[((tapBase + kw) * COUT + m) * CIN] + kfrag;
                const float* lB1 = lB0 + 16 * CIN;
                #pragma unroll
                for (int ks = 0; ks < 4; ++ks) {
                    int c4 = ks << 2;                  // constant offsets below
                    v2f b0 = *(const v2f*)(lB0 + c4);
                    v2f b1 = *(const v2f*)(lB1 + c4);
                    v2f a0 = *(const v2f*)(pa + c4);
                    v2f a1 = *(const v2f*)(pa + 256 + c4);
                    v2f a2 = *(const v2f*)(pa + 512 + c4);
                    v2f a3 = *(const v2f*)(pa + 768 + c4);
                    acc[0][0] = __builtin_amdgcn_wmma_f32_16x16x4_f32(false, a0, false, b0, (short)0, acc[0][0], false, false);
                    acc[1][0] = __builtin_amdgcn_wmma_f32_16x16x4_f32(false, a0, false, b1, (short)0, acc[1][0], false, false);
                    acc[0][1] = __builtin_amdgcn_wmma_f32_16x16x4_f32(false, a1, false, b0, (short)0, acc[0][1], false, false);
                    acc[1][1] = __builtin_amdgcn_wmma_f32_16x16x4_f32(false, a1, false, b1, (short)0, acc[1][1], false, false);
                    acc[0][2] = __builtin_amdgcn_wmma_f32_16x16x4_f32(false, a2, false, b0, (short)0, acc[0][2], false, false);
                    acc[1][2] = __builtin_amdgcn_wmma_f32_16x16x4_f32(false, a2, false, b1, (short)0, acc[1][2], false, false);
                    acc[0][3] = __builtin_amdgcn_wmma_f32_16x16x4_f32(false, a3, false, b0, (short)0, acc[0][3], false, false);
                    acc[1][3] = __builtin_amdgcn_wmma_f32_16x16x4_f32(false, a3, false, b1, (short)0, acc[1][3], false, false);
                }
            }
        }
    }

    // bias + per-channel partial stats (lane's values share couts m, 16+m)
    float bv0 = bias[m];
    float bv1 = bias[16 + m];
    float s0 = 0.f, q0 = 0.f, s1 = 0.f, q1 = 0.f;
    #pragma unroll
    for (int t = 0; t < 4; ++t)
        #pragma unroll
        for (int i = 0; i < 8; ++i) {
            acc[0][t][i] += bv0; acc[1][t][i] += bv1;
            s0 += acc[0][t][i]; q0 += acc[0][t][i] * acc[0][t][i];
            s1 += acc[1][t][i]; q1 += acc[1][t][i] * acc[1][t][i];
        }
    atomicAdd(&lSum[m],             s0);
    atomicAdd(&lSum[COUT + m],      q0);
    atomicAdd(&lSum[16 + m],        s1);
    atomicAdd(&lSum[COUT + 16 + m], q1);

    // store: lane holds D[M = v + 8*hi][N]; w = 16*t + 8*hi + v contiguous
    int wb = hi << 3;
    float* o0 = y + ((((n * COUT + m)      * DD + d) * HH + h) << 6) + wb;
    float* o1 = y + ((((n * COUT + 16 + m) * DD + d) * HH + h) << 6) + wb;
    #pragma unroll
    for (int t = 0; t < 4; ++t) {
        *(float4*)(o0 + 16 * t)     = make_float4(acc[0][t][0], acc[0][t][1], acc[0][t][2], acc[0][t][3]);
        *(float4*)(o0 + 16 * t + 4) = make_float4(acc[0][t][4], acc[0][t][5], acc[0][t][6], acc[0][t][7]);
        *(float4*)(o1 + 16 * t)     = make_float4(acc[1][t][0], acc[1][t][1], acc[1][t][2], acc[1][t][3]);
        *(float4*)(o1 + 16 * t + 4) = make_float4(acc[1][t][4], acc[1][t][5], acc[1][t][6], acc[1][t][7]);
    }

    __syncthreads();
    if (tid < 2 * COUT) atomicAdd(&g_sums[tid], lSum[tid]);
}

// ---------------------------------------------------------------------------
// Kernel 5: BatchNorm (batch stats, biased var, eps=1e-5) + softplus, in place
// float4 per thread; a block stays within one (n,cout) slab -> scalar stats
// ---------------------------------------------------------------------------
__global__ __launch_bounds__(256) void finalize_bn_softplus(float* __restrict__ y,
                                                            const float* __restrict__ g_sums,
                                                            const float* __restrict__ gamma,
                                                            const float* __restrict__ beta) {
    int e = (blockIdx.x * 256 + threadIdx.x) << 2;   // NOUT/4 threads
    int cout = (e >> 16) & 31;                       // 65536 px per (n,cout) slab
    const float inv_cnt = 1.0f / (float)NPIX;
    float mean = g_sums[cout] * inv_cnt;
    float var  = g_sums[COUT + cout] * inv_cnt - mean * mean;
    float sc = rsqrtf(var + 1e-5f) * gamma[cout];
    float sh = beta[cout] - mean * sc;
    float4 v = *(float4*)(y + e);
    float rr[4] = {v.x, v.y, v.z, v.w};
    #pragma unroll
    for (int i = 0; i < 4; ++i) {
        float t = rr[i] * sc + sh;
        rr[i] = (t > 20.0f) ? t : log1pf(expf(t));
    }
    *(float4*)(y + e) = make_float4(rr[0], rr[1], rr[2], rr[3]);
}

// ---------------------------------------------------------------------------
// Host-side launcher
// ---------------------------------------------------------------------------
extern "C" void kernel_launch(void* const* d_in, const int* in_sizes, int n_in,
                              void* d_out, int out_size, void* d_ws, size_t ws_size,
                              hipStream_t stream) {
    const float* x     = (const float*)d_in[0];
    const float* w     = (const float*)d_in[1];
    const float* bias  = (const float*)d_in[2];
    const float* gamma = (const float*)d_in[3];
    const float* beta  = (const float*)d_in[4];
    float* out = (float*)d_out;

    char* ws = (char*)d_ws;
    float* Wp   = (float*)ws;                       // 55,296 B
    float* sums = (float*)(ws + 55296);             // 256 B
    float* xT   = (float*)(ws + 55552);             // 4096*1056*4 = 17,301,504 B

    prep_weights<<<1, 512, 0, stream>>>(w, Wp);
    zero_sums<<<1, 64, 0, stream>>>(sums);
    transpose_x<<<NROW, 256, 0, stream>>>(x, xT);
    conv_wmma<<<CONV_BLOCKS, 256, 0, stream>>>(xT, Wp, bias, out, sums);
    finalize_bn_softplus<<<NOUT / 1024, 256, 0, stream>>>(out, sums, gamma, beta);
}